// BiDAF_71777493451047
// MI455X (gfx1250) — compile-verified
//
#include <hip/hip_runtime.h>

// Problem constants from the reference
#define NN 16
#define TT 2048
#define JJ 2048
#define DD 256

#define LDSPITCH 260   // 256 + 4 pad floats: row stride 1040B -> bank-conflict-free
                       // fragment reads (260 mod 64 = 4) and 16B-aligned rows (1040=65*16)
#define RTILES 4       // row-tiles of 16 per block (amortizes W staging)

typedef __attribute__((ext_vector_type(2))) float v2f;
typedef __attribute__((ext_vector_type(8))) float v8f;

// ---------------------------------------------------------------------------
// Kernel A: s[row] = relu(X[row,:] @ W^T + b) . wd
// 512 threads = 16 waves. W (256x256) and the 16-row A tile are staged in LDS
// (coalesced b128 global loads), then each wave runs 64 x
// V_WMMA_F32_16X16X4_F32 fed by conflict-free ds_load_b64 fragment reads.
// Dynamic LDS: 256*260 floats (W) + 16*260 floats (A) = 282,880 B (< 320KB WGP).
// ---------------------------------------------------------------------------
__global__ __launch_bounds__(512) void proj_dot_kernel(
    const float* __restrict__ X,     // (R, 256) rows
    const float* __restrict__ W,     // (256, 256), W[e,d]
    const float* __restrict__ bvec,  // (256)
    const float* __restrict__ wd,    // (256) slice of w_dot
    float* __restrict__ sOut)        // (R)
{
    extern __shared__ float lds[];
    float* Wl = lds;                     // 256 * LDSPITCH
    float* Al = lds + 256 * LDSPITCH;    // 16  * LDSPITCH
    __shared__ float lds_s[16];

    const int tid    = threadIdx.x;
    const int wave   = tid >> 5;         // 0..15
    const int lane   = tid & 31;
    const int lane16 = lane & 15;
    const int hi     = lane >> 4;        // 0/1: K sub-offset within fragment

    // ---- stage W into padded LDS (coalesced b128 -> ds_store_b128) ----
    for (int f = tid; f < 256 * 64; f += 512) {        // 16K float4s
        const int row = f >> 6;
        const int c4  = (f & 63) << 2;
        float4 w4 = *(const float4*)(W + row * DD + c4);
        *(float4*)(Wl + row * LDSPITCH + c4) = w4;
    }
    __syncthreads();

    const int e = wave * 16 + lane16;    // output feature this lane owns in B
    const float be = bvec[e];
    const float we = wd[e];
    const float* bFrag = Wl + e * LDSPITCH + hi * 2;
    const float* aFrag = Al + lane16 * LDSPITCH + hi * 2;

    for (int tile = 0; tile < RTILES; ++tile) {
        const int rowBase = (blockIdx.x * RTILES + tile) * 16;

        // ---- stage 16-row A tile ----
        for (int f = tid; f < 16 * 64; f += 512) {     // 1K float4s
            const int row = f >> 6;
            const int c4  = (f & 63) << 2;
            float4 a4 = *(const float4*)(X + (size_t)(rowBase + row) * DD + c4);
            *(float4*)(Al + row * LDSPITCH + c4) = a4;
        }
        if (tid < 16) lds_s[tid] = 0.0f;
        __syncthreads();

        // ---- K=256 accumulation: 64 WMMAs fed from LDS ----
        v8f acc = {};
#pragma unroll 8
        for (int k0 = 0; k0 < DD; k0 += 4) {
            v2f a  = *(const v2f*)(aFrag + k0);        // ds_load_b64, no conflicts
            v2f bb = *(const v2f*)(bFrag + k0);
            acc = __builtin_amdgcn_wmma_f32_16x16x4_f32(
                false, a, false, bb, (short)0, acc, false, false);
        }

        // ---- ReLU + dot with wd, reduce 16 e-lanes -> per-row scalar ----
#pragma unroll
        for (int v = 0; v < 8; ++v) {
            float y = acc[v] + be;
            y = y > 0.0f ? y : 0.0f;
            float contrib = y * we;
            contrib += __shfl_xor(contrib, 1, 32);     // stays inside each 16-lane half
            contrib += __shfl_xor(contrib, 2, 32);
            contrib += __shfl_xor(contrib, 4, 32);
            contrib += __shfl_xor(contrib, 8, 32);
            if (lane16 == 0) {
                atomicAdd(&lds_s[v + 8 * hi], contrib);  // ds_add_f32
            }
        }
        __syncthreads();
        if (tid < 16) sOut[rowBase + tid] = lds_s[tid];
        __syncthreads();   // protect Al / lds_s before next tile
    }
}

// ---------------------------------------------------------------------------
// Kernel B: per-n masked softmax over sU (t-independent: sH + b0 cancels in
// the renormalized masked softmax). Also zeroes v[n,:].
// ---------------------------------------------------------------------------
__global__ __launch_bounds__(256) void softmax_kernel(
    const float* __restrict__ sU,    // (N, J)
    const int* __restrict__ U_mask,  // (N, J)
    float* __restrict__ p,           // (N, J) out
    float* __restrict__ v)           // (N, D) zeroed here
{
    __shared__ float red[256];
    const int n   = blockIdx.x;
    const int tid = threadIdx.x;
    const float* s = sU + n * JJ;
    const int*   m = U_mask + n * JJ;

    float mx = -3.4e38f;
    for (int j = tid; j < JJ; j += 256) {
        float val = m[j] ? s[j] : -3.4e38f;
        mx = fmaxf(mx, val);
    }
    red[tid] = mx; __syncthreads();
    for (int off = 128; off > 0; off >>= 1) {
        if (tid < off) red[tid] = fmaxf(red[tid], red[tid + off]);
        __syncthreads();
    }
    mx = red[0]; __syncthreads();

    float sum = 0.0f;
    for (int j = tid; j < JJ; j += 256) {
        float e = m[j] ? __expf(s[j] - mx) : 0.0f;
        p[n * JJ + j] = e;
        sum += e;
    }
    red[tid] = sum; __syncthreads();
    for (int off = 128; off > 0; off >>= 1) {
        if (tid < off) red[tid] += red[tid + off];
        __syncthreads();
    }
    const float inv = 1.0f / (red[0] + 1e-13f);
    for (int j = tid; j < JJ; j += 256) {
        p[n * JJ + j] *= inv;
    }
    v[n * DD + tid] = 0.0f;
}

// ---------------------------------------------------------------------------
// Kernel B2: v[n,d] = sum_j p[n,j] * U[n,j,d]; 16 j-chunks per n merged with
// global_atomic_add_f32.
// ---------------------------------------------------------------------------
__global__ __launch_bounds__(256) void attend_kernel(
    const float* __restrict__ p,  // (N,J)
    const float* __restrict__ U,  // (N,J,D)
    float* __restrict__ v)        // (N,D) accumulate
{
    const int n  = blockIdx.y;
    const int jb = blockIdx.x;    // 16 chunks of 128 j
    const int d  = threadIdx.x;   // 256
    const int j0 = jb * 128;
    const float* pp = p + n * JJ + j0;
    const float* uu = U + ((size_t)n * JJ + j0) * DD + d;
    float accv = 0.0f;
#pragma unroll 4
    for (int j = 0; j < 128; ++j) {
        accv += pp[j] * uu[(size_t)j * DD];
    }
    atomicAdd(&v[n * DD + d], accv);
}

// ---------------------------------------------------------------------------
// Kernel C: pure-bandwidth writer.
//   S[n,t,j]   = sH[n,t] + sU[n,j] + b0      (268 MB, b128 stores)
//   c2q[n,t,:] = v[n,:]                      (33.5 MB, broadcast over t)
// ---------------------------------------------------------------------------
__global__ __launch_bounds__(256) void output_kernel(
    const float* __restrict__ sH,     // (N,T)
    const float* __restrict__ sU,     // (N,J)
    const float* __restrict__ v,      // (N,D)
    const float* __restrict__ b_dot,  // (1)
    float* __restrict__ c2q,          // (N,T,D)
    float* __restrict__ S)            // (N,T,J)
{
    __shared__ float lsU[JJ];
    __shared__ float lv[DD];
    const int tid = threadIdx.x;
    const int blk = blockIdx.x;          // 0..1023
    const int n   = blk >> 6;            // / (T/32)
    const int t0  = (blk & 63) * 32;
    const float b0 = b_dot[0];

#pragma unroll
    for (int i = 0; i < 8; ++i)
        lsU[tid + i * 256] = sU[n * JJ + tid + i * 256] + b0;
    lv[tid] = v[n * DD + tid];
    __syncthreads();

    for (int t = t0; t < t0 + 32; ++t) {
        const float hv = sH[n * TT + t];
        float* Srow = S + (size_t)(n * TT + t) * JJ;
#pragma unroll
        for (int jc = 0; jc < 2; ++jc) {
            const int j = jc * 1024 + tid * 4;
            float4 o;
            o.x = lsU[j + 0] + hv;
            o.y = lsU[j + 1] + hv;
            o.z = lsU[j + 2] + hv;
            o.w = lsU[j + 3] + hv;
            *(float4*)(Srow + j) = o;    // global_store_b128
        }
        c2q[(size_t)(n * TT + t) * DD + tid] = lv[tid];
    }
}

// ---------------------------------------------------------------------------
extern "C" void kernel_launch(void* const* d_in, const int* in_sizes, int n_in,
                              void* d_out, int out_size, void* d_ws, size_t ws_size,
                              hipStream_t stream) {
    const float* U      = (const float*)d_in[0];  // (16,2048,256)
    const float* H      = (const float*)d_in[1];  // (16,2048,256)
    const int*   U_mask = (const int*)  d_in[2];  // (16,2048)
    // d_in[3] = H_mask: unused by the reference math
    const float* W      = (const float*)d_in[4];  // (256,256)
    const float* b      = (const float*)d_in[5];  // (256)
    const float* w_dot  = (const float*)d_in[6];  // (512)
    const float* b_dot  = (const float*)d_in[7];  // (1)

    float* out = (float*)d_out;
    float* c2q = out;                                // N*T*D floats
    float* S   = out + (size_t)NN * TT * DD;         // N*T*J floats

    float* ws = (float*)d_ws;
    float* sH = ws;                  // N*T   = 32768
    float* sU = sH + NN * TT;        // N*J   = 32768
    float* p  = sU + NN * JJ;        // N*J   = 32768
    float* v  = p  + NN * JJ;        // N*D   = 4096

    const size_t projLds = (size_t)(256 + 16) * LDSPITCH * sizeof(float); // 282,880 B

    // Projection + dot: 512 blocks each (32768 rows / (16 rows * RTILES))
    proj_dot_kernel<<<(NN * TT) / (16 * RTILES), 512, projLds, stream>>>(
        H, W, b, w_dot, sH);
    proj_dot_kernel<<<(NN * JJ) / (16 * RTILES), 512, projLds, stream>>>(
        U, W, b, w_dot + DD, sU);

    softmax_kernel<<<NN, 256, 0, stream>>>(sU, U_mask, p, v);

    dim3 g2(16, NN);
    attend_kernel<<<g2, 256, 0, stream>>>(p, U, v);

    output_kernel<<<(NN * TT) / 32, 256, 0, stream>>>(sH, sU, v, b_dot, c2q, S);
}